// get_loss_pre_19061064860389
// MI455X (gfx1250) — compile-verified
//
#include <hip/hip_runtime.h>
#include <hip/hip_bf16.h>

// Chamfer-distance loss via V_WMMA_F32_16X16X4_F32 (CDNA5 / gfx1250, wave32).
//
// Augmented-GEMM trick: with A row = (qx,qy,qz,1) and B col = (-2tx,-2ty,-2tz,|t|^2),
// one K=4 f32 WMMA yields c[m][n] = |t_n|^2 - 2<q_m,t_n>. Running min over n in the
// loop; |q|^2 add and max(.,0) clamp commute with min -> applied once in the tail.
// Two template instantiations (one per chamfer direction) give literal trip counts
// (scalar loop control, EXEC always all-1s as WMMA requires). Tail min-reduction
// across lanes uses ds_swizzle_b32 xor patterns. Deterministic fixed-order final sum.

typedef __attribute__((ext_vector_type(2))) float v2f;
typedef __attribute__((ext_vector_type(8))) float v8f;

#if defined(__HIP_DEVICE_COMPILE__)
#if !__has_builtin(__builtin_amdgcn_wmma_f32_16x16x4_f32)
#error "gfx1250 device pass: __builtin_amdgcn_wmma_f32_16x16x4_f32 not available"
#endif
#endif

namespace {
constexpr int kB = 4;
constexpr int kN = 8192;   // shape points per batch
constexpr int kM = 2048;   // skel points per batch
constexpr int kTiles1 = kB * (kN / 16);          // 2048
constexpr int kTiles2 = kB * (kM / 16);          // 512
constexpr int kTilesTotal = kTiles1 + kTiles2;   // 2560
constexpr int kWavesPerBlock = 8;                // 256 threads, wave32
}

// min with the lane at (lane ^ XOR) via ds_swizzle_b32 group-of-32 mode:
// offset = {1'b0, xor[4:0], or[4:0]=0, and[4:0]=0x1f}
template <int XOR>
__device__ __forceinline__ float swz_min(float v) {
  const int o = __builtin_amdgcn_ds_swizzle(__float_as_int(v), 0x1f | (XOR << 10));
  return fminf(v, __int_as_float(o));
}
template <int XOR>
__device__ __forceinline__ float swz_add(float v) {
  const int o = __builtin_amdgcn_ds_swizzle(__float_as_int(v), 0x1f | (XOR << 10));
  return v + __int_as_float(o);
}

// QN: query points per batch, TN: target points per batch.
template <int QN, int TN>
__global__ __launch_bounds__(256) void chamfer_wmma_kernel(
    const float* __restrict__ qryb,   // [kB, QN, 3]
    const float* __restrict__ tgtb,   // [kB, TN, 3]
    float* __restrict__ wsSeg)        // [kB * QN/16] partial sums
{
  constexpr int kQTilesPerBatch = QN / 16;
  constexpr int kTT = TN / 16;       // literal inner trip count

  const int lane = threadIdx.x & 31;
  const int wave = threadIdx.x >> 5;
  const int tile = blockIdx.x * kWavesPerBlock + wave;   // wave-uniform

  const int b = tile / kQTilesPerBatch;
  const int t = tile % kQTilesPerBatch;
  const float* qry = qryb + (size_t)b * QN * 3;
  const float* tgt = tgtb + (size_t)b * TN * 3;
  const int qbase = t * 16;

  const int half = lane >> 4;   // 0: holds K=0,1  1: holds K=2,3
  const int l16 = lane & 15;    // row (A) / column (B,C,D) within tile

  // ---- augmented A fragment: row = (qx, qy, qz, 1) ----
  const float* qp = qry + (size_t)(qbase + l16) * 3;
  const float qx = qp[0], qy = qp[1], qz = qp[2];
  v2f afrag;
  afrag.x = half ? qz : qx;     // K=0 (x) | K=2 (z)
  afrag.y = half ? 1.0f : qy;   // K=1 (y) | K=3 (1)

  float mn[8];
#pragma unroll
  for (int i = 0; i < 8; ++i) mn[i] = 3.4e38f;

#pragma unroll 4
  for (int jt = 0; jt < kTT; ++jt) {
    // ---- augmented B fragment: col = (-2tx, -2ty, -2tz, |t|^2) ----
    const float* tp = tgt + (size_t)(jt * 16 + l16) * 3;
    const float tx = tp[0], ty = tp[1], tz = tp[2];
    const float sq2 = fmaf(tx, tx, fmaf(ty, ty, tz * tz));
    v2f bfrag;
    bfrag.x = -2.0f * (half ? tz : tx);       // K=0 (-2x) | K=2 (-2z)
    bfrag.y = half ? sq2 : (-2.0f * ty);      // K=1 (-2y) | K=3 (|t|^2)

    v8f c = {};
    c = __builtin_amdgcn_wmma_f32_16x16x4_f32(
        /*neg_a=*/false, afrag, /*neg_b=*/false, bfrag,
        /*c_mod=*/(short)0, c, /*reuse_a=*/false, /*reuse_b=*/false);

    // c[i] = |t_col|^2 - 2<q_row, t_col> ; running min over columns
#pragma unroll
    for (int i = 0; i < 8; ++i) mn[i] = fminf(mn[i], c[i]);
  }

  // ---- tail: min across the 16 columns of each half, add |q|^2, clamp, sum ----
  // C/D layout: VGPR i -> row qbase+i (lanes 0-15) / qbase+i+8 (lanes 16-31)
  float sum = 0.0f;
#pragma unroll
  for (int i = 0; i < 8; ++i) {
    float v = mn[i];
    v = swz_min<1>(v);
    v = swz_min<2>(v);
    v = swz_min<4>(v);
    v = swz_min<8>(v);
    const float* p = qry + (size_t)(qbase + i + 8 * half) * 3;
    const float sq1 = fmaf(p[0], p[0], fmaf(p[1], p[1], p[2] * p[2]));
    sum += fmaxf(sq1 + v, 0.0f);
  }
  // lane 0: rows qbase+0..7 ; lane 16: rows qbase+8..15
  sum = swz_add<16>(sum);
  if (lane == 0) wsSeg[tile] = sum;
}

__global__ __launch_bounds__(256) void chamfer_reduce_kernel(
    const float* __restrict__ ws, float* __restrict__ out)
{
  __shared__ float s[256];
  float acc = 0.0f;
  for (int i = threadIdx.x; i < kTilesTotal; i += 256) acc += ws[i];
  s[threadIdx.x] = acc;
  __syncthreads();
  for (int off = 128; off > 0; off >>= 1) {
    if ((int)threadIdx.x < off) s[threadIdx.x] += s[threadIdx.x + off];
    __syncthreads();
  }
  if (threadIdx.x == 0) out[0] = s[0] * 1.0e-4f;
}

extern "C" void kernel_launch(void* const* d_in, const int* in_sizes, int n_in,
                              void* d_out, int out_size, void* d_ws, size_t ws_size,
                              hipStream_t stream) {
  const float* shape = (const float*)d_in[0];  // [4, 8192, 3] fp32
  const float* skel  = (const float*)d_in[1];  // [4, 2048, 3] fp32
  float* out = (float*)d_out;                  // scalar fp32
  float* ws  = (float*)d_ws;                   // >= kTilesTotal floats (10 KB)

  // dir 1: every shape point -> nearest skel point (2048 tiles = 256 blocks)
  chamfer_wmma_kernel<kN, kM>
      <<<kTiles1 / kWavesPerBlock, 256, 0, stream>>>(shape, skel, ws);
  // dir 2: every skel point -> nearest shape point (512 tiles = 64 blocks)
  chamfer_wmma_kernel<kM, kN>
      <<<kTiles2 / kWavesPerBlock, 256, 0, stream>>>(skel, shape, ws + kTiles1);

  chamfer_reduce_kernel<<<1, 256, 0, stream>>>(ws, out);
}